// Model_60610578481304
// MI455X (gfx1250) — compile-verified
//
#include <hip/hip_runtime.h>

typedef __attribute__((ext_vector_type(2))) float v2f;
typedef __attribute__((ext_vector_type(8))) float v8f;

#define BATCH   64
#define CIN     256
#define COUT    256
#define HWSZ    1024      // 32*32
#define KTAPS   16        // 4*4
#define INV_AREA (1.0f / 1225.0f)   // 1/(35*35)

// ---------------------------------------------------------------------------
// Pass 1: Sx[b,ci] = sum over 32x32 spatial. One wave32 per (b,ci) row.
// Row = 1024 contiguous floats = 256 float4; lane reads 8 float4's.
// ---------------------------------------------------------------------------
__global__ __launch_bounds__(256) void reduce_x_kernel(
    const float* __restrict__ x, float* __restrict__ sx) {
  const int tid  = blockIdx.x * blockDim.x + threadIdx.x;
  const int row  = tid >> 5;          // 0 .. 16383  == b*256 + ci
  const int lane = threadIdx.x & 31;

  const float4* p = (const float4*)(x + (size_t)row * HWSZ);
  float s = 0.0f;
#pragma unroll
  for (int i = 0; i < 8; ++i) {
    float4 v = p[lane + 32 * i];
    s += (v.x + v.y) + (v.z + v.w);
  }
#pragma unroll
  for (int off = 16; off > 0; off >>= 1) s += __shfl_xor(s, off, 32);
  if (lane == 0) sx[row] = s;
}

// ---------------------------------------------------------------------------
// Pass 2: Sw[ci,co] = sum over 16 kernel taps. One lane per (ci,co) pair,
// 16 contiguous floats = 4 float4 loads.
// ---------------------------------------------------------------------------
__global__ __launch_bounds__(256) void reduce_w_kernel(
    const float* __restrict__ w, float* __restrict__ sw) {
  const int t = blockIdx.x * blockDim.x + threadIdx.x;   // 0 .. 65535 = ci*256+co
  const float4* p = (const float4*)(w + (size_t)t * KTAPS);
  float4 a = p[0], b = p[1], c = p[2], d = p[3];
  sw[t] = ((a.x + a.y) + (a.z + a.w)) + ((b.x + b.y) + (b.z + b.w)) +
          ((c.x + c.y) + (c.z + c.w)) + ((d.x + d.y) + (d.z + d.w));
}

// ---------------------------------------------------------------------------
// Pass 3: out[b,co] = (Sx[b,:] dot Sw[:,co]) * INV_AREA + bias[co]
// M=64, N=256, K=256 fp32 GEMM via V_WMMA_F32_16X16X4_F32.
// One wave per 16x16 tile: 4 x 16 = 64 tiles = 8 blocks x 8 waves.
// ---------------------------------------------------------------------------
__global__ __launch_bounds__(256) void gemm_wmma_kernel(
    const float* __restrict__ sx, const float* __restrict__ sw,
    const float* __restrict__ bias, float* __restrict__ out) {
  const int lane  = threadIdx.x & 31;
  const int wave  = (blockIdx.x * blockDim.x + threadIdx.x) >> 5;  // 0..63
  const int mt    = wave >> 4;     // 0..3
  const int nt    = wave & 15;     // 0..15
  const int m0    = mt * 16;
  const int n0    = nt * 16;
  const int lmod  = lane & 15;
  const int khalf = (lane >> 4) * 2;   // 0 or 2

  const float* arow = sx + (size_t)(m0 + lmod) * CIN;  // A row (this lane's M)
  const float* bcol = sw + (n0 + lmod);                // B column (this lane's N)

  v8f acc = {};
  for (int k = 0; k < CIN; k += 4) {
    // A[M=lmod+m0][k+khalf .. k+khalf+1] : contiguous float2
    v2f a = *(const v2f*)(arow + k + khalf);
    // B[k+khalf .. +1][N=lmod+n0] : stride-COUT column loads
    v2f b;
    b.x = bcol[(size_t)(k + khalf) * COUT];
    b.y = bcol[(size_t)(k + khalf + 1) * COUT];
    acc = __builtin_amdgcn_wmma_f32_16x16x4_f32(
        /*neg_a=*/false, a, /*neg_b=*/false, b,
        /*c_mod=*/(short)0, acc, /*reuse_a=*/false, /*reuse_b=*/false);
  }

  // C/D layout: VGPR r -> lanes 0-15: (M=r, N=lane); lanes 16-31: (M=r+8, N=lane-16)
  const int n     = n0 + lmod;
  const float bv  = bias[n];
  const int mbase = m0 + (lane >> 4) * 8;
#pragma unroll
  for (int r = 0; r < 8; ++r) {
    out[(size_t)(mbase + r) * COUT + n] = acc[r] * INV_AREA + bv;
  }
}

// ---------------------------------------------------------------------------
// Launch
// ---------------------------------------------------------------------------
extern "C" void kernel_launch(void* const* d_in, const int* in_sizes, int n_in,
                              void* d_out, int out_size, void* d_ws, size_t ws_size,
                              hipStream_t stream) {
  const float* x    = (const float*)d_in[0];   // [64,256,32,32]
  const float* w    = (const float*)d_in[1];   // [256,256,4,4]
  const float* bias = (const float*)d_in[2];   // [256]
  float* out = (float*)d_out;                  // [64,256,1,1] = 16384 floats

  float* sx = (float*)d_ws;                    // 64*256   = 16384 floats (64 KB)
  float* sw = sx + BATCH * CIN;                // 256*256  = 65536 floats (256 KB)

  // Pass 1: 16384 waves, 8 waves/block -> 2048 blocks
  reduce_x_kernel<<<(BATCH * CIN) / 8, 256, 0, stream>>>(x, sx);
  // Pass 2: 65536 lanes -> 256 blocks
  reduce_w_kernel<<<(CIN * COUT) / 256, 256, 0, stream>>>(w, sw);
  // Pass 3: 64 waves -> 8 blocks of 8 waves
  gemm_wmma_kernel<<<8, 256, 0, stream>>>(sx, sw, bias, out);
}